// GraphGPSLayer_33775622816118
// MI455X (gfx1250) — compile-verified
//
#include <hip/hip_runtime.h>

// ---------------------------------------------------------------------------
// GraphGPS layer for MI455X (gfx1250). fp32 end-to-end; matrix ops via
// V_WMMA_F32_16X16X4_F32 (wave32, 16x16 tiles, K=4 per instruction).
// GEMM tiles staged to LDS with async global->LDS copies (ASYNCcnt) and
// double buffering when the toolchain exposes the gfx1250 async builtins.
// ---------------------------------------------------------------------------

typedef float v2f __attribute__((ext_vector_type(2)));
typedef float v8f __attribute__((ext_vector_type(8)));
typedef int   v4i __attribute__((ext_vector_type(4)));

constexpr int HD    = 256;  // hidden size
constexpr int NHEAD = 8;
constexpr int DHEAD = 32;   // HD / NHEAD
constexpr int MMAX  = 128;  // max nodes per graph (dense slot count)

#if defined(__gfx1250__) &&                                        \
    __has_builtin(__builtin_amdgcn_global_load_async_to_lds_b128) && \
    __has_builtin(__builtin_amdgcn_s_wait_asynccnt)
#define USE_ASYNC_LDS 1
#else
#define USE_ASYNC_LDS 0
#endif

__device__ __forceinline__ v8f wmma_f32(v2f a, v2f b, v8f c) {
  // 8 args: (neg_a, A, neg_b, B, c_mod, C, reuse_a, reuse_b)
  return __builtin_amdgcn_wmma_f32_16x16x4_f32(false, a, false, b, (short)0, c,
                                               false, false);
}

#if USE_ASYNC_LDS
// async b128 copy: global (AS1) -> LDS (AS3); tracked by ASYNCcnt.
__device__ __forceinline__ void async_copy_b128(const float* g, float* l) {
  __builtin_amdgcn_global_load_async_to_lds_b128(
      (v4i __attribute__((address_space(1)))*)g,
      (v4i __attribute__((address_space(3)))*)l, 0, 0);
}
#endif

// ---------------------------------------------------------------------------
// cum[b] = prefix sum of batch_num_nodes (B is tiny: one thread)
// ---------------------------------------------------------------------------
__global__ void cum_kernel(const int* __restrict__ bnn, int* __restrict__ cum,
                           int B) {
  if (blockIdx.x == 0 && threadIdx.x == 0) {
    int acc = 0;
    for (int i = 0; i < B; ++i) { cum[i] = acc; acc += bnn[i]; }
    cum[B] = acc;
  }
}

// ---------------------------------------------------------------------------
// z = h  (float4 grid-stride copy)
// ---------------------------------------------------------------------------
__global__ void copy_kernel(const float4* __restrict__ src,
                            float4* __restrict__ dst, int n4) {
  int stride = gridDim.x * blockDim.x;
  for (int i = blockIdx.x * blockDim.x + threadIdx.x; i < n4; i += stride)
    dst[i] = src[i];
}

// ---------------------------------------------------------------------------
// z[dst] += h[src] over edges; one block per edge, thread = channel.
// ---------------------------------------------------------------------------
__global__ __launch_bounds__(HD) void edge_agg_kernel(
    const float* __restrict__ h, const int* __restrict__ esrc,
    const int* __restrict__ edst, float* __restrict__ z, int E) {
  int e = blockIdx.x;
  if (e >= E) return;
  int s = esrc[e];
  int d = edst[e];
  int c = threadIdx.x;
  atomicAdd(&z[(size_t)d * HD + c], h[(size_t)s * HD + c]);
}

// ---------------------------------------------------------------------------
// Y = act( X[NxKin] * W[KoutxKin]^T + bias [+ residual] )
// grid: (Nrows/128, Kout/64); 256 threads = 8 waves; wave w -> 16-row strip,
// 4 output tiles of 16x16; K staged in LDS 32 at a time, double-buffered,
// next chunk prefetched with GLOBAL_LOAD_ASYNC_TO_LDS_B128 while WMMAs run.
// ---------------------------------------------------------------------------
__global__ __launch_bounds__(256) void gemm_wmma_kernel(
    const float* __restrict__ X, const float* __restrict__ W,
    const float* __restrict__ bias, const float* __restrict__ resid,
    float* __restrict__ Y, int Nrows, int Kin, int Kout, int relu) {
  // 36-float row stride: 144 B -> every row 16B-aligned for async b128
  // destinations; 36*lane mod 64 banks distinct across lanes 0..15.
  __shared__ float Xs[2][128][36];
  __shared__ float Ws[2][64][36];

  const int tid  = threadIdx.x;
  const int wave = tid >> 5;
  const int lane = tid & 31;
  const int ll   = lane & 15;
  const int hi   = lane >> 4;  // 0: K pair (kk,kk+1), 1: (kk+2,kk+3)
  const int rowBase = blockIdx.x * 128;
  const int colBase = blockIdx.y * 64;

  auto stage = [&](int bufi, int k0) {
#if USE_ASYNC_LDS
    // 128x32 X tile: 1024 b128 transfers; 64x32 W tile: 512 transfers.
    for (int i = tid; i < 128 * 8; i += 256) {
      int r = i >> 3, c4 = (i & 7) << 2;
      async_copy_b128(&X[(size_t)(rowBase + r) * Kin + k0 + c4],
                      &Xs[bufi][r][c4]);
    }
    for (int i = tid; i < 64 * 8; i += 256) {
      int r = i >> 3, c4 = (i & 7) << 2;
      async_copy_b128(&W[(size_t)(colBase + r) * Kin + k0 + c4],
                      &Ws[bufi][r][c4]);
    }
#else
    for (int i = tid; i < 128 * 32; i += 256) {
      int r = i >> 5, c = i & 31;
      Xs[bufi][r][c] = X[(size_t)(rowBase + r) * Kin + k0 + c];
    }
    for (int i = tid; i < 64 * 32; i += 256) {
      int r = i >> 5, c = i & 31;
      Ws[bufi][r][c] = W[(size_t)(colBase + r) * Kin + k0 + c];
    }
#endif
  };

  v8f acc[4];
#pragma unroll
  for (int j = 0; j < 4; ++j)
#pragma unroll
    for (int r = 0; r < 8; ++r) acc[j][r] = 0.0f;

  const int arow = wave * 16 + ll;

  stage(0, 0);
#if USE_ASYNC_LDS
  __builtin_amdgcn_s_wait_asynccnt(0);
#endif
  __syncthreads();

  int bufi = 0;
  for (int k0 = 0; k0 < Kin; k0 += 32) {
    const bool more = (k0 + 32) < Kin;
    if (more) stage(bufi ^ 1, k0 + 32);  // async prefetch next K chunk

#pragma unroll
    for (int kk = 0; kk < 32; kk += 4) {
      const int ka = kk + hi * 2;
      v2f a = *(const v2f*)&Xs[bufi][arow][ka];
#pragma unroll
      for (int jt = 0; jt < 4; ++jt) {
        v2f b = *(const v2f*)&Ws[bufi][jt * 16 + ll][ka];
        acc[jt] = wmma_f32(a, b, acc[jt]);
      }
    }

    if (more) {
#if USE_ASYNC_LDS
      __builtin_amdgcn_s_wait_asynccnt(0);
#endif
      __syncthreads();
      bufi ^= 1;
    }
  }

  // Epilogue. D layout: VGPR r -> (M = hi*8 + r, N = ll)
  const int mBase = rowBase + wave * 16 + hi * 8;
#pragma unroll
  for (int jt = 0; jt < 4; ++jt) {
    const int col = colBase + jt * 16 + ll;
    const float bv = bias[col];
#pragma unroll
    for (int r = 0; r < 8; ++r) {
      const int row = mBase + r;
      float v = acc[jt][r] + bv;
      if (resid) v += resid[(size_t)row * Kout + col];
      if (relu) v = v > 0.0f ? v : 0.0f;
      Y[(size_t)row * Kout + col] = v;
    }
  }
}

// ---------------------------------------------------------------------------
// Multi-head self-attention: one workgroup per (graph, head).
// qkv: [N x 3H] rows in node order; graph b occupies rows cum[b]..cum[b]+nb.
// ao:  [N x H] attention output per node (pre out-proj).
// ---------------------------------------------------------------------------
__global__ __launch_bounds__(256) void attn_kernel(
    const float* __restrict__ qkv, const int* __restrict__ cum,
    const int* __restrict__ bnn, float* __restrict__ ao) {
  __shared__ float Qs[MMAX][34];
  __shared__ float Ks[MMAX][34];
  __shared__ float Vt[DHEAD][130];   // transposed V: Vt[d][slot]
  __shared__ float S[MMAX][130];     // scores / probabilities

  const int b    = blockIdx.x / NHEAD;
  const int head = blockIdx.x % NHEAD;
  const int tid  = threadIdx.x;
  const int wave = tid >> 5;
  const int lane = tid & 31;
  const int ll   = lane & 15;
  const int hi   = lane >> 4;

  const int n0 = cum[b];
  const int nb = bnn[b];

  // Stage Q, K, V^T for this (graph, head); invalid slots -> 0.
  for (int i = tid; i < MMAX * DHEAD; i += 256) {
    int s = i >> 5, d = i & 31;
    float q = 0.0f, k = 0.0f, v = 0.0f;
    if (s < nb) {
      size_t base = (size_t)(n0 + s) * (3 * HD) + head * DHEAD + d;
      q = qkv[base];
      k = qkv[base + HD];
      v = qkv[base + 2 * HD];
    }
    Qs[s][d] = q;
    Ks[s][d] = k;
    Vt[d][s] = v;
  }
  __syncthreads();

  const float scale = 0.17677669529663687f;  // 1/sqrt(32)
  const int arow = wave * 16 + ll;

  // S = (Q K^T) * scale ; wave w computes rows [16w, 16w+16)
  for (int jt = 0; jt < 8; ++jt) {
    v8f acc;
#pragma unroll
    for (int r = 0; r < 8; ++r) acc[r] = 0.0f;
#pragma unroll
    for (int kk = 0; kk < DHEAD; kk += 4) {
      const int ka = kk + hi * 2;
      v2f a = *(const v2f*)&Qs[arow][ka];
      v2f bf = *(const v2f*)&Ks[jt * 16 + ll][ka];
      acc = wmma_f32(a, bf, acc);
    }
    const int m0 = wave * 16 + hi * 8;
#pragma unroll
    for (int r = 0; r < 8; ++r) S[m0 + r][jt * 16 + ll] = acc[r] * scale;
  }
  __syncthreads();

  // Row-wise masked softmax (keys >= nb excluded, probability set to 0).
  if (tid < MMAX) {
    const int r = tid;
    float mx = -1e30f;
    for (int c = 0; c < nb; ++c) mx = fmaxf(mx, S[r][c]);
    float sum = 0.0f;
    for (int c = 0; c < nb; ++c) {
      float e = __expf(S[r][c] - mx);
      S[r][c] = e;
      sum += e;
    }
    const float inv = sum > 0.0f ? 1.0f / sum : 0.0f;
    for (int c = 0; c < nb; ++c) S[r][c] *= inv;
    for (int c = nb; c < MMAX; ++c) S[r][c] = 0.0f;
  }
  __syncthreads();

  // O = P * V ; wave w: rows [16w,16w+16), two 16-col tiles, K = 128 slots.
  for (int jt = 0; jt < 2; ++jt) {
    v8f acc;
#pragma unroll
    for (int r = 0; r < 8; ++r) acc[r] = 0.0f;
#pragma unroll
    for (int kk = 0; kk < MMAX; kk += 4) {
      const int ka = kk + hi * 2;
      v2f a = *(const v2f*)&S[arow][ka];
      v2f bf = *(const v2f*)&Vt[jt * 16 + ll][ka];
      acc = wmma_f32(a, bf, acc);
    }
    const int m0 = wave * 16 + hi * 8;
#pragma unroll
    for (int r = 0; r < 8; ++r) {
      const int slot = m0 + r;
      if (slot < nb) {
        const int node = n0 + slot;
        ao[(size_t)node * HD + head * DHEAD + jt * 16 + ll] = acc[r];
      }
    }
  }
}

// ---------------------------------------------------------------------------
// BatchNorm stats: per-channel sum & sumsq over all rows (thread = channel).
// stats[c] += sum, stats[HD + c] += sumsq  (caller zeroes stats first).
// ---------------------------------------------------------------------------
__global__ __launch_bounds__(HD) void bn_stats_kernel(
    const float* __restrict__ X, float* __restrict__ stats, int Nrows) {
  const int c = threadIdx.x;
  const int rows_per = (Nrows + gridDim.x - 1) / gridDim.x;
  const int r0 = blockIdx.x * rows_per;
  const int r1 = min(r0 + rows_per, Nrows);
  float s = 0.0f, s2 = 0.0f;
  for (int r = r0; r < r1; ++r) {
    float x = X[(size_t)r * HD + c];
    s += x;
    s2 += x * x;
  }
  atomicAdd(&stats[c], s);
  atomicAdd(&stats[HD + c], s2);
}

// hh = BN(local_pre) + BN(attn_pre)
__global__ void bn_combine_kernel(
    const float* __restrict__ xl, const float* __restrict__ xa,
    const float* __restrict__ stl, const float* __restrict__ sta,
    const float* __restrict__ gl, const float* __restrict__ bl,
    const float* __restrict__ ga, const float* __restrict__ ba,
    float* __restrict__ out, int total, float invN) {
  const int stride = gridDim.x * blockDim.x;
  for (int i = blockIdx.x * blockDim.x + threadIdx.x; i < total; i += stride) {
    const int c = i & (HD - 1);
    const float ml = stl[c] * invN;
    const float vl = stl[HD + c] * invN - ml * ml;
    const float sl = gl[c] * rsqrtf(vl + 1e-5f);
    const float ma = sta[c] * invN;
    const float va = sta[HD + c] * invN - ma * ma;
    const float sa = ga[c] * rsqrtf(va + 1e-5f);
    out[i] = (xl[i] - ml) * sl + bl[c] + (xa[i] - ma) * sa + ba[c];
  }
}

// out = BN(x)
__global__ void bn_apply_kernel(const float* __restrict__ x,
                                const float* __restrict__ st,
                                const float* __restrict__ g,
                                const float* __restrict__ bb,
                                float* __restrict__ out, int total,
                                float invN) {
  const int stride = gridDim.x * blockDim.x;
  for (int i = blockIdx.x * blockDim.x + threadIdx.x; i < total; i += stride) {
    const int c = i & (HD - 1);
    const float m = st[c] * invN;
    const float v = st[HD + c] * invN - m * m;
    const float s = g[c] * rsqrtf(v + 1e-5f);
    out[i] = (x[i] - m) * s + bb[c];
  }
}

// ---------------------------------------------------------------------------
// Host-side orchestration
// ---------------------------------------------------------------------------
extern "C" void kernel_launch(void* const* d_in, const int* in_sizes, int n_in,
                              void* d_out, int out_size, void* d_ws,
                              size_t ws_size, hipStream_t stream) {
  const float* h         = (const float*)d_in[0];
  const float* gin_w1    = (const float*)d_in[1];
  const float* gin_b1    = (const float*)d_in[2];
  const float* gin_w2    = (const float*)d_in[3];
  const float* gin_b2    = (const float*)d_in[4];
  const float* in_proj_w = (const float*)d_in[5];
  const float* in_proj_b = (const float*)d_in[6];
  const float* out_proj_w= (const float*)d_in[7];
  const float* out_proj_b= (const float*)d_in[8];
  const float* bn_local_g= (const float*)d_in[9];
  const float* bn_local_b= (const float*)d_in[10];
  const float* bn_attn_g = (const float*)d_in[11];
  const float* bn_attn_b = (const float*)d_in[12];
  const float* bn_out_g  = (const float*)d_in[13];
  const float* bn_out_b  = (const float*)d_in[14];
  const float* ffn1_w    = (const float*)d_in[15];
  const float* ffn1_b    = (const float*)d_in[16];
  const float* ffn2_w    = (const float*)d_in[17];
  const float* ffn2_b    = (const float*)d_in[18];
  const int*   esrc      = (const int*)d_in[19];
  const int*   edst      = (const int*)d_in[20];
  const int*   bnn       = (const int*)d_in[21];

  const int Nrows = in_sizes[0] / HD;   // 32768
  const int E     = in_sizes[19];       // 524288
  const int Bg    = in_sizes[21];       // 256
  const int total = Nrows * HD;
  const float invN = 1.0f / (float)Nrows;

  // Workspace carve-up
  float* buf0  = (float*)d_ws;                       // N*H   (z -> ao)
  float* buf1  = buf0 + (size_t)Nrows * HD;          // N*H   (y1 -> attn_pre)
  float* buf2  = buf1 + (size_t)Nrows * HD;          // N*H   (local_pre -> out_pre)
  float* buf3  = buf2 + (size_t)Nrows * HD;          // N*3H  (qkv -> hh + ffn1_out)
  float* stats = buf3 + (size_t)Nrows * 3 * HD;      // 6*H floats (3 BN sets)
  int*   cum   = (int*)(stats + 6 * HD);             // B+1 ints
  float* ffn1o = buf3 + (size_t)Nrows * HD;          // N*2H, aliases tail of buf3

  (void)hipMemsetAsync(stats, 0, 6 * HD * sizeof(float), stream);
  cum_kernel<<<1, 1, 0, stream>>>(bnn, cum, Bg);

  // z = h + segment_sum(h[src] -> dst)
  copy_kernel<<<2048, 256, 0, stream>>>((const float4*)h, (float4*)buf0,
                                        total / 4);
  edge_agg_kernel<<<E, HD, 0, stream>>>(h, esrc, edst, buf0, E);

  // GIN MLP
  gemm_wmma_kernel<<<dim3(Nrows / 128, HD / 64), 256, 0, stream>>>(
      buf0, gin_w1, gin_b1, nullptr, buf1, Nrows, HD, HD, 1);
  gemm_wmma_kernel<<<dim3(Nrows / 128, HD / 64), 256, 0, stream>>>(
      buf1, gin_w2, gin_b2, h, buf2, Nrows, HD, HD, 0);  // local_pre = h + MLP(z)
  bn_stats_kernel<<<128, HD, 0, stream>>>(buf2, stats, Nrows);

  // QKV projection (on node rows; dense slots for graph b are rows
  // cum[b]..cum[b]+nb, so no scatter is needed)
  gemm_wmma_kernel<<<dim3(Nrows / 128, (3 * HD) / 64), 256, 0, stream>>>(
      h, in_proj_w, in_proj_b, nullptr, buf3, Nrows, HD, 3 * HD, 0);

  // Attention per (graph, head)
  attn_kernel<<<Bg * NHEAD, 256, 0, stream>>>(buf3, cum, bnn, buf0);

  // Out projection + residual
  gemm_wmma_kernel<<<dim3(Nrows / 128, HD / 64), 256, 0, stream>>>(
      buf0, out_proj_w, out_proj_b, h, buf1, Nrows, HD, HD, 0);
  bn_stats_kernel<<<128, HD, 0, stream>>>(buf1, stats + 2 * HD, Nrows);

  // hh = BN(local_pre) + BN(attn_pre)
  bn_combine_kernel<<<2048, 256, 0, stream>>>(
      buf2, buf1, stats, stats + 2 * HD, bn_local_g, bn_local_b, bn_attn_g,
      bn_attn_b, buf3, total, invN);

  // FFN
  gemm_wmma_kernel<<<dim3(Nrows / 128, (2 * HD) / 64), 256, 0, stream>>>(
      buf3, ffn1_w, ffn1_b, nullptr, ffn1o, Nrows, HD, 2 * HD, 1);
  gemm_wmma_kernel<<<dim3(Nrows / 128, HD / 64), 256, 0, stream>>>(
      ffn1o, ffn2_w, ffn2_b, buf3, buf2, Nrows, 2 * HD, HD, 0);  // out_pre

  // Final BN -> d_out
  bn_stats_kernel<<<128, HD, 0, stream>>>(buf2, stats + 4 * HD, Nrows);
  bn_apply_kernel<<<2048, 256, 0, stream>>>(buf2, stats + 4 * HD, bn_out_g,
                                            bn_out_b, (float*)d_out, total,
                                            invN);
}